// ColorExtractor_64742337019920
// MI455X (gfx1250) — compile-verified
//
#include <hip/hip_runtime.h>
#include <stddef.h>

typedef float v2f __attribute__((ext_vector_type(2)));
typedef float v4f __attribute__((ext_vector_type(4)));
typedef float v8f __attribute__((ext_vector_type(8)));

#define NB        64            // batches
#define NPIX      (512 * 512)   // pixels per batch
#define KC        8             // clusters
#define ITERS     10
#define THREADS   256           // 8 wave32 waves
#define TILES     32            // 32-pixel tiles per wave per block
#define PPB       (THREADS * TILES)      // 8192 pixels per block
#define BLK_PER_B (NPIX / PPB)           // 32 blocks per batch
#define SLOT      36            // floats per per-thread LDS slot (8*4 + pad), 144B

// ws layout (floats): [0, NB*KC*4)           centroids (x,y,z,pad)
//                     [NB*KC*4, 2*NB*KC*4)   sums (sx,sy,sz,count)
#define CENT_F    (NB * KC * 4)

__device__ __forceinline__ unsigned hash_u32(unsigned v) {
    v ^= v >> 16; v *= 0x7feb352du;
    v ^= v >> 15; v *= 0x846ca68bu;
    v ^= v >> 16;
    return v;
}

// lane l <-> lane l^16 swap: v_permlanex16_b32 (VALU) instead of ds_bpermute (LDS)
__device__ __forceinline__ float swap16(float v) {
#if __has_builtin(__builtin_amdgcn_permlanex16)
    int i = __builtin_amdgcn_permlanex16(__float_as_int(v), __float_as_int(v),
                                         0x76543210, 0xFEDCBA98, false, false);
    return __int_as_float(i);
#else
    return __shfl_xor(v, 16);
#endif
}

// one 32-pixel tile: build B operands, 2x WMMA score, per-lane argmin, LDS RMW
__device__ __forceinline__ void tile_step(float x, float y, float z,
                                          bool lo, v2f A, v8f cq, v4f* myslot4) {
    // build B operands (4x16 f32, cols = pixels) via VALU half-swap
    float xx = swap16(x);
    float yy = swap16(y);
    float zz = swap16(z);
    v2f B1, B2;
    B1.x = lo ? x  : zz;  B1.y = lo ? y  : 0.0f;  // cols = pixels base..base+15
    B2.x = lo ? xx : z;   B2.y = lo ? yy : 0.0f;  // cols = pixels base+16..+31

    // D = (-2C)xB + q  ==>  D[k] = |c_k|^2 - 2 p.c_k  (the score directly)
    v8f d1 = __builtin_amdgcn_wmma_f32_16x16x4_f32(
        false, A, false, B1, (short)0, cq, false, false);
    v8f d2 = __builtin_amdgcn_wmma_f32_16x16x4_f32(
        false, A, false, B2, (short)0, cq, false, false);

    // lanes 0-15: C1 rows 0..7 for col=lane (my pixel)
    // lanes 16-31: C2 rows 8..15 (dup clusters) for col=lane-16 (my pixel)
    // argmin with raw selects (scores are finite; skip fminf NaN-canonicalize)
    int   best = 0;
    float bs   = lo ? d1[0] : d2[0];
#pragma unroll
    for (int k = 1; k < KC; ++k) {
        float s    = lo ? d1[k] : d2[k];
        bool  less = s < bs;
        best = less ? k : best;
        bs   = less ? s : bs;
    }

    // private-slot accumulate: ds_load_b128 + ds_store_b128 per pixel
    v4f* cell = myslot4 + best;
    v4f  v    = *cell;
    v.x += x; v.y += y; v.z += z; v.w += 1.0f;
    *cell = v;
}

// ---------------- init: seed centroids from hashed pixel picks, zero sums ----------
__global__ void km_init(const float* __restrict__ px, float* __restrict__ ws) {
    int tid = blockIdx.x * blockDim.x + threadIdx.x;
    float* cent = ws;
    float* sums = ws + CENT_F;
    if (tid < CENT_F) sums[tid] = 0.0f;
    if (tid < NB * KC) {
        int b = tid >> 3, k = tid & 7;
        unsigned idx = hash_u32(0x9E3779B9u * (unsigned)b + 0x85EBCA6Bu * (unsigned)k + 42u) % NPIX;
        const float* p = px + ((size_t)b * NPIX + idx) * 3;
        cent[tid * 4 + 0] = p[0];
        cent[tid * 4 + 1] = p[1];
        cent[tid * 4 + 2] = p[2];
        cent[tid * 4 + 3] = 0.0f;
    }
}

// ---------------- assignment + partial sums (WMMA core) ---------------------------
__global__ void __launch_bounds__(THREADS)
km_assign(const float* __restrict__ px, float* __restrict__ ws) {
    const float* cent = ws;
    float*       sums = ws + CENT_F;

    const int b    = blockIdx.y;
    const int lane = threadIdx.x & 31;
    const int w    = threadIdx.x >> 5;

    __shared__ float acc[THREADS * SLOT];   // 36 KB

    // zero my private slot (16B-aligned, stride 144B)
    v4f* myslot4 = (v4f*)&acc[threadIdx.x * SLOT];
#pragma unroll
    for (int i = 0; i < SLOT / 4; ++i) myslot4[i] = (v4f)(0.0f);

    // centroids + |c|^2
    const v4f* cb = (const v4f*)(cent + b * KC * 4);
    v4f  c[KC];
    v8f  cq;                       // WMMA C operand: score bias q_k per row
#pragma unroll
    for (int k = 0; k < KC; ++k) {
        c[k] = cb[k];
        cq[k] = c[k].x * c[k].x + c[k].y * c[k].y + c[k].z * c[k].z;
    }

    // A operand (16x4 f32) holds -2*centroid: rows 0..7 = clusters, rows 8..15 dup.
    // lanes 0-15: (K0,K1)=(-2cx,-2cy); lanes 16-31: (K2,K3)=(-2cz,0)
    const bool lo = lane < 16;
    const int  kk = lane & 7;
    v2f A;
    A.x = -2.0f * (lo ? c[kk].x : c[kk].z);
    A.y = lo ? (-2.0f * c[kk].y) : 0.0f;

    const float* pb   = px + (size_t)b * NPIX * 3;
    const int    base = blockIdx.x * PPB + w * 32 + lane;

    // software-pipelined: unconditional prefetch, last tile peeled out of the loop
    float x, y, z;
    {
        const float* pp = pb + (size_t)base * 3;
        x = pp[0]; y = pp[1]; z = pp[2];
    }
    for (int t = 0; t < TILES - 1; ++t) {
        const float* pn = pb + (size_t)(base + (t + 1) * THREADS) * 3;
        float nx = pn[0], ny = pn[1], nz = pn[2];    // global_load_b96 (prefetch)
        tile_step(x, y, z, lo, A, cq, myslot4);
        x = nx; y = ny; z = nz;
    }
    tile_step(x, y, z, lo, A, cq, myslot4);          // last tile, no prefetch

    __syncthreads();

    // tree-fold 256 slots down to slot 0 (32 useful floats each)
    for (int s = THREADS / 2; s > 0; s >>= 1) {
        if (threadIdx.x < (unsigned)s) {
            v4f* a0 = (v4f*)&acc[threadIdx.x * SLOT];
            v4f* a1 = (v4f*)&acc[(threadIdx.x + s) * SLOT];
#pragma unroll
            for (int i = 0; i < 8; ++i) a0[i] += a1[i];
        }
        __syncthreads();
    }

    if (threadIdx.x < 32) {
        atomicAdd(&sums[b * KC * 4 + threadIdx.x], acc[threadIdx.x]);
    }
}

// ---------------- update: centroid = sum/count (if count>0), zero sums -------------
__global__ void km_update(float* __restrict__ ws) {
    int tid = blockIdx.x * blockDim.x + threadIdx.x;
    if (tid >= NB * KC) return;
    float* cent = ws;
    float* sums = ws + CENT_F;
    float sx  = sums[tid * 4 + 0];
    float sy  = sums[tid * 4 + 1];
    float sz  = sums[tid * 4 + 2];
    float cnt = sums[tid * 4 + 3];
    if (cnt > 0.0f) {
        float inv = 1.0f / cnt;
        cent[tid * 4 + 0] = sx * inv;
        cent[tid * 4 + 1] = sy * inv;
        cent[tid * 4 + 2] = sz * inv;
    }
    sums[tid * 4 + 0] = 0.0f;
    sums[tid * 4 + 1] = 0.0f;
    sums[tid * 4 + 2] = 0.0f;
    sums[tid * 4 + 3] = 0.0f;
}

// ---------------- finalize: strip padding -> d_out (64 x 24) ----------------------
__global__ void km_finalize(const float* __restrict__ ws, float* __restrict__ out) {
    int tid = blockIdx.x * blockDim.x + threadIdx.x;
    if (tid >= NB * KC * 3) return;
    int b = tid / 24, r = tid % 24, k = r / 3, d = r % 3;
    out[tid] = ws[(b * KC + k) * 4 + d];
}

extern "C" void kernel_launch(void* const* d_in, const int* in_sizes, int n_in,
                              void* d_out, int out_size, void* d_ws, size_t ws_size,
                              hipStream_t stream) {
    const float* px  = (const float*)d_in[0];
    float*       ws  = (float*)d_ws;
    float*       out = (float*)d_out;

    km_init<<<(CENT_F + 255) / 256, 256, 0, stream>>>(px, ws);
    for (int it = 0; it < ITERS; ++it) {
        km_assign<<<dim3(BLK_PER_B, NB), THREADS, 0, stream>>>(px, ws);
        km_update<<<(NB * KC + 255) / 256, 256, 0, stream>>>(ws);
    }
    km_finalize<<<(NB * KC * 3 + 255) / 256, 256, 0, stream>>>(ws, out);
}